// CostModel_11527692222734
// MI455X (gfx1250) — compile-verified
//
#include <hip/hip_runtime.h>
#include <math.h>

typedef __attribute__((ext_vector_type(2))) float v2f;
typedef __attribute__((ext_vector_type(8))) float v8f;

#define HIDDIM 128
#define NHEAD  4
#define CH     32

__device__ __forceinline__ void atomicMaxFloat(float* addr, float val) {
    // signed-magnitude trick: max(float) == max(int) for >=0, min(uint) for <0
    if (val >= 0.0f) atomicMax((int*)addr, __float_as_int(val));
    else             atomicMin((unsigned int*)addr, __float_as_uint(val));
}

__global__ void fill_kernel(float* __restrict__ p, float v, size_t n) {
    size_t i = (size_t)blockIdx.x * blockDim.x + threadIdx.x;
    if (i < n) p[i] = v;
}

// Y[M,128] = act(X[M,K] @ W[K,128] (+ b)); M multiple of 16, K in {64,128}.
// 256 threads = 8 waves; wave w computes the 16x16 tile at columns [16w,16w+16).
// fp32 WMMA: V_WMMA_F32_16X16X4_F32, K stepped by 4.
template<int K, bool BIAS, bool RELU>
__global__ __launch_bounds__(256)
void wmma_gemm(const float* __restrict__ X, const float* __restrict__ W,
               const float* __restrict__ b, float* __restrict__ Y) {
    __shared__ float As[16 * K];
    const int m0   = blockIdx.x * 16;
    const int tid  = threadIdx.x;
    const int wave = tid >> 5;
    const int lane = tid & 31;
    const int n0   = wave * 16;

    // cooperative, coalesced load of the 16xK A tile
    for (int i = tid; i < 16 * K; i += 256) {
        int r = i / K, k = i - r * K;
        As[i] = X[(size_t)(m0 + r) * K + k];
    }
    __syncthreads();

    const int row = lane & 15;
    const int hi  = lane >> 4;   // 0: lanes 0-15, 1: lanes 16-31
    v8f acc = {0.f, 0.f, 0.f, 0.f, 0.f, 0.f, 0.f, 0.f};

    for (int k = 0; k < K; k += 4) {
        // A 16x4 fp32 layout: VGPR0 = {K=k   (lo lanes), K=k+2 (hi lanes)}
        //                     VGPR1 = {K=k+1 (lo lanes), K=k+3 (hi lanes)}
        v2f a, bb;
        a.x = As[row * K + k + 2 * hi + 0];
        a.y = As[row * K + k + 2 * hi + 1];
        // B 4x16 fp32 layout: VGPR0 = {row k (lo), row k+2 (hi)}, cols = n0+lane%16
        bb.x = W[(size_t)(k + 2 * hi + 0) * HIDDIM + n0 + row];
        bb.y = W[(size_t)(k + 2 * hi + 1) * HIDDIM + n0 + row];
        acc = __builtin_amdgcn_wmma_f32_16x16x4_f32(
            false, a, false, bb, (short)0, acc, false, false);
    }

    // C/D layout: VGPR j -> row m0+j (lo lanes) / m0+j+8 (hi lanes), col n0+lane%16
#pragma unroll
    for (int j = 0; j < 8; ++j) {
        int r = m0 + j + hi * 8;
        int c = n0 + row;
        float v = acc[j];
        if (BIAS) v += b[c];
        if (RELU) v = fmaxf(v, 0.0f);
        Y[(size_t)r * HIDDIM + c] = v;
    }
}

// per-(node,head) attention scores: asrc/adst[n*4+h] = dot(t[n, h*32:(h+1)*32], att)
__global__ void attn_scores(const float* __restrict__ t,
                            const float* __restrict__ a_src,
                            const float* __restrict__ a_dst,
                            float* __restrict__ asrc, float* __restrict__ adst,
                            int Nn) {
    int idx = blockIdx.x * blockDim.x + threadIdx.x;  // n*NHEAD + h
    if (idx >= Nn * NHEAD) return;
    int h = idx & (NHEAD - 1);
    const float* tv = t + (size_t)idx * CH;           // == t + n*128 + h*32
    float s1 = 0.f, s2 = 0.f;
#pragma unroll 8
    for (int c = 0; c < CH; ++c) {
        float v = tv[c];
        s1 += v * a_src[h * CH + c];
        s2 += v * a_dst[h * CH + c];
    }
    asrc[idx] = s1;
    adst[idx] = s2;
}

__device__ __forceinline__ void edge_sd(const int* __restrict__ ei, size_t j,
                                        int E, int& s, int& d) {
    if (j < (size_t)E) { s = ei[j]; d = ei[(size_t)E + j]; }
    else               { s = d = (int)(j - E); }   // appended self loops
}

__global__ void edge_max(const int* __restrict__ ei, int E, int Nn,
                         const float* __restrict__ asrc, const float* __restrict__ adst,
                         float* __restrict__ m) {
    size_t idx = (size_t)blockIdx.x * blockDim.x + threadIdx.x;
    size_t EP = (size_t)E + Nn;
    if (idx >= EP * NHEAD) return;
    int h = (int)(idx & (NHEAD - 1));
    size_t j = idx >> 2;
    int s, d; edge_sd(ei, j, E, s, d);
    float e = asrc[(size_t)s * NHEAD + h] + adst[(size_t)d * NHEAD + h];
    e = (e > 0.f) ? e : 0.2f * e;                  // leaky_relu(0.2)
    atomicMaxFloat(&m[(size_t)d * NHEAD + h], e);
}

__global__ void edge_sumexp(const int* __restrict__ ei, int E, int Nn,
                            const float* __restrict__ asrc, const float* __restrict__ adst,
                            const float* __restrict__ m, float* __restrict__ z) {
    size_t idx = (size_t)blockIdx.x * blockDim.x + threadIdx.x;
    size_t EP = (size_t)E + Nn;
    if (idx >= EP * NHEAD) return;
    int h = (int)(idx & (NHEAD - 1));
    size_t j = idx >> 2;
    int s, d; edge_sd(ei, j, E, s, d);
    float e = asrc[(size_t)s * NHEAD + h] + adst[(size_t)d * NHEAD + h];
    e = (e > 0.f) ? e : 0.2f * e;
    atomicAdd(&z[(size_t)d * NHEAD + h], expf(e - m[(size_t)d * NHEAD + h]));
}

__global__ void edge_aggr(const int* __restrict__ ei, int E, int Nn,
                          const float* __restrict__ asrc, const float* __restrict__ adst,
                          const float* __restrict__ m, const float* __restrict__ z,
                          const float* __restrict__ t, float* __restrict__ out) {
    size_t idx = (size_t)blockIdx.x * blockDim.x + threadIdx.x;
    size_t EP = (size_t)E + Nn;
    if (idx >= EP * NHEAD) return;
    int h = (int)(idx & (NHEAD - 1));
    size_t j = idx >> 2;
    int s, d; edge_sd(ei, j, E, s, d);
    float e = asrc[(size_t)s * NHEAD + h] + adst[(size_t)d * NHEAD + h];
    e = (e > 0.f) ? e : 0.2f * e;
    float alpha = expf(e - m[(size_t)d * NHEAD + h]) /
                  (z[(size_t)d * NHEAD + h] + 1e-16f);
    const float* ts = t   + (size_t)s * HIDDIM + h * CH;
    float*       od = out + (size_t)d * HIDDIM + h * CH;
#pragma unroll 8
    for (int c = 0; c < CH; ++c) atomicAdd(&od[c], alpha * ts[c]);
}

__global__ void bias_relu(float* __restrict__ hbuf, const float* __restrict__ bconv,
                          size_t total) {
    size_t i = (size_t)blockIdx.x * blockDim.x + threadIdx.x;
    if (i < total) hbuf[i] = fmaxf(hbuf[i] + bconv[i & (HIDDIM - 1)], 0.0f);
}

__global__ void pool_nodes(const float* __restrict__ hbuf, const int* __restrict__ batch,
                           float* __restrict__ psum, float* __restrict__ pmax,
                           float* __restrict__ cnt, int Nn) {
    size_t idx = (size_t)blockIdx.x * blockDim.x + threadIdx.x;
    if (idx >= (size_t)Nn * HIDDIM) return;
    int n = (int)(idx >> 7), i = (int)(idx & (HIDDIM - 1));
    int g = batch[n];
    float v = hbuf[idx];
    atomicAdd(&psum[(size_t)g * HIDDIM + i], v);
    atomicMaxFloat(&pmax[(size_t)g * HIDDIM + i], v);
    if (i == 0) atomicAdd(&cnt[g], 1.0f);
}

__global__ void pool_finalize(const float* __restrict__ psum, const float* __restrict__ pmax,
                              const float* __restrict__ cnt, float* __restrict__ p, int G) {
    int idx = blockIdx.x * blockDim.x + threadIdx.x;
    if (idx >= G * 2 * HIDDIM) return;
    int g = idx >> 8, i = idx & (2 * HIDDIM - 1);
    float c = cnt[g];
    if (i < HIDDIM) {
        p[idx] = psum[g * HIDDIM + i] / fmaxf(c, 1.0f);
    } else {
        float mx = pmax[g * HIDDIM + (i - HIDDIM)];
        p[idx] = (c > 0.0f) ? mx : 0.0f;
    }
}

// one block per graph; p[G,256] -> 256 -> 128 -> 1
__global__ __launch_bounds__(256)
void mlp_head(const float* __restrict__ p,
              const float* __restrict__ W1, const float* __restrict__ b1,
              const float* __restrict__ W2, const float* __restrict__ b2,
              const float* __restrict__ W3, const float* __restrict__ b3,
              float* __restrict__ out) {
    __shared__ float s0[256], s1[256], s2[128];
    int g = blockIdx.x, t = threadIdx.x;
    s0[t] = p[g * 256 + t];
    __syncthreads();
    float a1 = b1[t];
    for (int k = 0; k < 256; ++k) a1 += s0[k] * W1[k * 256 + t];
    s1[t] = fmaxf(a1, 0.0f);
    __syncthreads();
    if (t < 128) {
        float a2 = b2[t];
        for (int k = 0; k < 256; ++k) a2 += s1[k] * W2[k * 128 + t];
        s2[t] = fmaxf(a2, 0.0f);
    }
    __syncthreads();
    if (t == 0) {
        float a3 = b3[0];
        for (int k = 0; k < 128; ++k) a3 += s2[k] * W3[k];
        out[g] = a3;
    }
}

static inline unsigned nblk(size_t n, int b) { return (unsigned)((n + b - 1) / b); }

extern "C" void kernel_launch(void* const* d_in, const int* in_sizes, int n_in,
                              void* d_out, int out_size, void* d_ws, size_t ws_size,
                              hipStream_t stream) {
    const float* x       = (const float*)d_in[0];
    const int*   ei      = (const int*)d_in[1];
    const int*   batch   = (const int*)d_in[2];
    const float* Wp      = (const float*)d_in[3];
    const float* bp      = (const float*)d_in[4];
    const float* Wl      = (const float*)d_in[5];   // [3,128,128]
    const float* att_src = (const float*)d_in[6];   // [3,4,32]
    const float* att_dst = (const float*)d_in[7];
    const float* bconv   = (const float*)d_in[8];   // [3,128]
    const float* W1 = (const float*)d_in[9];
    const float* b1 = (const float*)d_in[10];
    const float* W2 = (const float*)d_in[11];
    const float* b2 = (const float*)d_in[12];
    const float* W3 = (const float*)d_in[13];
    const float* b3 = (const float*)d_in[14];
    float* out = (float*)d_out;

    const int N = in_sizes[0] / 64;        // 50000
    const int E = in_sizes[1] / 2;         // 1600000
    const int G = out_size;                // 8
    const size_t NH = (size_t)N * HIDDIM;
    const size_t EP = (size_t)E + N;

    float* w     = (float*)d_ws;
    float* bufA  = w;                      // layer input / aggregation accumulator
    float* bufB  = w + NH;                 // transformed features t = h @ W
    float* asrc  = w + 2 * NH;
    float* adst  = asrc + (size_t)N * NHEAD;
    float* mbuf  = adst + (size_t)N * NHEAD;
    float* zbuf  = mbuf + (size_t)N * NHEAD;
    float* psum  = zbuf + (size_t)N * NHEAD;
    float* pmax  = psum + (size_t)G * HIDDIM;
    float* cnt   = pmax + (size_t)G * HIDDIM;
    float* pvec  = cnt + G;                // [G, 256]

    const int B = 256;

    // input projection: bufA = relu(x @ Wp + bp), fp32 WMMA
    wmma_gemm<64, true, true><<<N / 16, 256, 0, stream>>>(x, Wp, bp, bufA);

    for (int l = 0; l < 3; ++l) {
        const float* Wl_l = Wl + (size_t)l * HIDDIM * HIDDIM;
        // t = h @ W (no bias, no act)
        wmma_gemm<128, false, false><<<N / 16, 256, 0, stream>>>(bufA, Wl_l, nullptr, bufB);
        // attention scores per (node, head)
        attn_scores<<<nblk((size_t)N * NHEAD, B), B, 0, stream>>>(
            bufB, att_src + l * NHEAD * CH, att_dst + l * NHEAD * CH, asrc, adst, N);
        // init reductions; bufA (old features, now dead) becomes accumulator
        fill_kernel<<<nblk((size_t)N * NHEAD, B), B, 0, stream>>>(mbuf, -INFINITY, (size_t)N * NHEAD);
        fill_kernel<<<nblk((size_t)N * NHEAD, B), B, 0, stream>>>(zbuf, 0.0f, (size_t)N * NHEAD);
        fill_kernel<<<nblk(NH, B), B, 0, stream>>>(bufA, 0.0f, NH);
        // scatter softmax over dst + weighted aggregation
        edge_max   <<<nblk(EP * NHEAD, B), B, 0, stream>>>(ei, E, N, asrc, adst, mbuf);
        edge_sumexp<<<nblk(EP * NHEAD, B), B, 0, stream>>>(ei, E, N, asrc, adst, mbuf, zbuf);
        edge_aggr  <<<nblk(EP * NHEAD, B), B, 0, stream>>>(ei, E, N, asrc, adst, mbuf, zbuf, bufB, bufA);
        // h = relu(agg + bconv[l])
        bias_relu<<<nblk(NH, B), B, 0, stream>>>(bufA, bconv + l * HIDDIM, NH);
    }

    // pooling (mean + max per graph)
    fill_kernel<<<nblk((size_t)G * HIDDIM, B), B, 0, stream>>>(psum, 0.0f, (size_t)G * HIDDIM);
    fill_kernel<<<nblk((size_t)G * HIDDIM, B), B, 0, stream>>>(pmax, -INFINITY, (size_t)G * HIDDIM);
    fill_kernel<<<nblk((size_t)G, B), B, 0, stream>>>(cnt, 0.0f, (size_t)G);
    pool_nodes<<<nblk(NH, B), B, 0, stream>>>(bufA, batch, psum, pmax, cnt, N);
    pool_finalize<<<nblk((size_t)G * 2 * HIDDIM, B), B, 0, stream>>>(psum, pmax, cnt, pvec, G);

    // MLP head
    mlp_head<<<G, 256, 0, stream>>>(pvec, W1, b1, W2, b2, W3, b3, out);
}